// DeepMD_43198781063817
// MI455X (gfx1250) — compile-verified
//
#include <hip/hip_runtime.h>
#include <math.h>

typedef __attribute__((ext_vector_type(2))) float v2f;
typedef __attribute__((ext_vector_type(8))) float v8f;

#define BB 2
#define NN 2048
#define KK 96
#define NATOM (BB * NN)          // 4096
#define E1D 32
#define E2D 64
#define FD 256
#define DRD 1024                 // 16 * E2D
#define PI_F 3.14159265358979323846f

// ---------------------------------------------------------------------------
// f32 WMMA 16x16x4 helper (CDNA5 V_WMMA_F32_16X16X4_F32)
// A fragment (lane l): M = l&15, K = 2*(l>>4) + {0,1}
// B fragment (lane l): N = l&15, K = 2*(l>>4) + {0,1}
// D fragment (lane l): row = v + 8*(l>>4), col = l&15   (v = 0..7)
// ---------------------------------------------------------------------------
__device__ __forceinline__ v8f wmma4(v2f a, v2f b, v8f c) {
  return __builtin_amdgcn_wmma_f32_16x16x4_f32(
      /*neg_a=*/false, a, /*neg_b=*/false, b,
      /*c_mod=*/(short)0, c, /*reuse_a=*/false, /*reuse_b=*/false);
}

// ---------------------------------------------------------------------------
// Kernel 1: per-atom descriptor forward.  One wave (32 threads) per atom.
// ---------------------------------------------------------------------------
__global__ __launch_bounds__(32) void embed_fwd(
    const float* __restrict__ imdR,
    const float* __restrict__ ew0, const float* __restrict__ eb0,
    const float* __restrict__ ew1, const float* __restrict__ eb1,
    float* __restrict__ DRws) {
  __shared__ float Ri[KK * 4];
  __shared__ float Gs[KK * E2D];
  __shared__ float tA[16 * 4];
  __shared__ float tB[4 * E2D];
  __shared__ float ew0s[E1D], eb0s[E1D], eb1s[E2D];
  __shared__ float ew1s[E1D * E2D];

  const int atom = blockIdx.x;
  const int lane = threadIdx.x;
  const float* base = imdR + (size_t)atom * KK * 4;

  for (int e = lane; e < E1D; e += 32) { ew0s[e] = ew0[e]; eb0s[e] = eb0[e]; }
  for (int e = lane; e < E2D; e += 32) eb1s[e] = eb1[e];
  for (int e = lane; e < E1D * E2D; e += 32) ew1s[e] = ew1[e];
  __syncthreads();

  for (int k = lane; k < KK; k += 32) {
    float x = base[k * 4 + 0], y = base[k * 4 + 1], z = base[k * 4 + 2];
    float nb = base[k * 4 + 3];
    bool mask = nb > 0.0f;
    float dR2 = x * x + y * y + z * z;
    float safe = (dR2 > 0.0f) ? dR2 : 1.0f;
    float S = 0.0f;
    if (mask && dR2 < 10.0f) S = 1.0f / safe;
    if (dR2 >= 10.0f && dR2 < 25.0f)
      S += 0.5f * cosf(PI_F * (dR2 - 10.0f) / 15.0f) + 0.5f;
    float coef = mask ? (S / dR2) : 0.0f;
    Ri[k * 4 + 0] = S;
    Ri[k * 4 + 1] = coef * x;
    Ri[k * 4 + 2] = coef * y;
    Ri[k * 4 + 3] = coef * z;

    float h[E1D];
#pragma unroll
    for (int j = 0; j < E1D; ++j) h[j] = tanhf(S * ew0s[j] + eb0s[j]);
    for (int i = 0; i < E2D; ++i) {
      float acc = eb1s[i];
#pragma unroll
      for (int j = 0; j < E1D; ++j) acc += h[j] * ew1s[j * E2D + i];
      Gs[k * E2D + i] = tanhf(acc);
    }
  }
  __syncthreads();

  // tmpA[m][c] = sum_k G[k][m] * Ri[k][c]   (m < 16)
  for (int e = lane; e < 64; e += 32) {
    int m = e >> 2, c = e & 3;
    float acc = 0.0f;
    for (int k = 0; k < KK; ++k) acc += Gs[k * E2D + m] * Ri[k * 4 + c];
    tA[e] = acc;
  }
  // tmpB[c][i] = sum_k Ri[k][c] * G[k][i]
  for (int e = lane; e < 4 * E2D; e += 32) {
    int c = e >> 6, i = e & 63;
    float acc = 0.0f;
    for (int k = 0; k < KK; ++k) acc += Ri[k * 4 + c] * Gs[k * E2D + i];
    tB[e] = acc;
  }
  __syncthreads();

  // DR[m][i] = sum_c tmpA[m][c] * tmpB[c][i]
  float* DRrow = DRws + ((size_t)atom << 10);
  for (int e = lane; e < DRD; e += 32) {
    int m = e >> 6, i = e & 63;
    float acc = 0.0f;
#pragma unroll
    for (int c = 0; c < 4; ++c) acc += tA[m * 4 + c] * tB[c * E2D + i];
    DRrow[e] = acc;
  }
}

// ---------------------------------------------------------------------------
// Kernel 2: fitting net forward (WMMA).  16 atoms / 256-thread workgroup.
// ---------------------------------------------------------------------------
__global__ __launch_bounds__(256) void fit_fwd(
    const float* __restrict__ DRws,
    const float* __restrict__ fw0, const float* __restrict__ fb0,
    const float* __restrict__ fw1, const float* __restrict__ fb1,
    const float* __restrict__ fw2, const float* __restrict__ fb2,
    float* __restrict__ f1ws, float* __restrict__ f2ws,
    float* __restrict__ out) {
  __shared__ float f1s[16 * FD];
  __shared__ float f2s[16 * FD];
  const int atom0 = blockIdx.x * 16;
  const int lane = threadIdx.x & 31;
  const int w = threadIdx.x >> 5;
  const int m = lane & 15;               // row / col within tile
  const int kb = (lane >> 4) * 2;        // K sub-offset for A/B fragments

  // ---- layer 1: z1 = DR @ fw0 + fb0 ; f1 = tanh(z1) ----
  const float* Arow = DRws + (size_t)(atom0 + m) * DRD;
  for (int nt = w * 2; nt < w * 2 + 2; ++nt) {
    const int ncol = nt * 16 + m;
    v8f acc = {0.f, 0.f, 0.f, 0.f, 0.f, 0.f, 0.f, 0.f};
    for (int k0 = 0; k0 < DRD; k0 += 4) {
      v2f a, b;
      a.x = Arow[k0 + kb];
      a.y = Arow[k0 + kb + 1];
      b.x = fw0[(size_t)(k0 + kb) * FD + ncol];
      b.y = fw0[(size_t)(k0 + kb + 1) * FD + ncol];
      acc = wmma4(a, b, acc);
    }
    float bias = fb0[ncol];
#pragma unroll
    for (int v = 0; v < 8; ++v) {
      int row = v + 8 * (lane >> 4);
      float t = tanhf(acc[v] + bias);
      f1s[row * FD + ncol] = t;
      f1ws[(size_t)(atom0 + row) * FD + ncol] = t;
    }
  }
  __syncthreads();

  // ---- layer 2: f2 = tanh(f1 @ fw1 + fb1) ----
  for (int nt = w * 2; nt < w * 2 + 2; ++nt) {
    const int ncol = nt * 16 + m;
    v8f acc = {0.f, 0.f, 0.f, 0.f, 0.f, 0.f, 0.f, 0.f};
    for (int k0 = 0; k0 < FD; k0 += 4) {
      v2f a, b;
      a.x = f1s[m * FD + k0 + kb];
      a.y = f1s[m * FD + k0 + kb + 1];
      b.x = fw1[(size_t)(k0 + kb) * FD + ncol];
      b.y = fw1[(size_t)(k0 + kb + 1) * FD + ncol];
      acc = wmma4(a, b, acc);
    }
    float bias = fb1[ncol];
#pragma unroll
    for (int v = 0; v < 8; ++v) {
      int row = v + 8 * (lane >> 4);
      float t = tanhf(acc[v] + bias);
      f2s[row * FD + ncol] = t;
      f2ws[(size_t)(atom0 + row) * FD + ncol] = t;
    }
  }
  __syncthreads();

  // ---- layer 3: Ei = f2 @ fw2 + fb2 ; Etot atomic ----
  if (threadIdx.x < 16) {
    int row = threadIdx.x;
    float s = fb2[0];
    for (int j = 0; j < FD; ++j) s += f2s[row * FD + j] * fw2[j];
    int atom = atom0 + row;
    out[2 + atom] = s;                       // Ei
    atomicAdd(&out[atom >> 11], s);          // Etot[b], b = atom / NN
  }
}

// ---------------------------------------------------------------------------
// Kernel 3: fitting net backward (WMMA against transposed weights) -> gDR.
// ---------------------------------------------------------------------------
__global__ __launch_bounds__(256) void fit_bwd(
    const float* __restrict__ f1ws, const float* __restrict__ f2ws,
    const float* __restrict__ fw0, const float* __restrict__ fw1,
    const float* __restrict__ fw2,
    float* __restrict__ gDRws) {
  __shared__ float gz2s[16 * FD];
  __shared__ float gz1s[16 * FD];
  const int atom0 = blockIdx.x * 16;
  const int lane = threadIdx.x & 31;
  const int w = threadIdx.x >> 5;
  const int m = lane & 15;
  const int kb = (lane >> 4) * 2;

  // gz2 = fw2 * (1 - f2^2)     (d(sum Ei)/dEi = 1)
  for (int e = threadIdx.x; e < 16 * FD; e += 256) {
    int mm = e >> 8, j = e & 255;
    float f2v = f2ws[(size_t)(atom0 + mm) * FD + j];
    gz2s[e] = fw2[j] * (1.0f - f2v * f2v);
  }
  __syncthreads();

  // g1 = gz2 @ fw1^T ; gz1 = g1 * (1 - f1^2)
  for (int nt = w * 2; nt < w * 2 + 2; ++nt) {
    const int ncol = nt * 16 + m;
    v8f acc = {0.f, 0.f, 0.f, 0.f, 0.f, 0.f, 0.f, 0.f};
    for (int k0 = 0; k0 < FD; k0 += 4) {
      v2f a, b;
      a.x = gz2s[m * FD + k0 + kb];
      a.y = gz2s[m * FD + k0 + kb + 1];
      b.x = fw1[(size_t)ncol * FD + k0 + kb];      // fw1^T
      b.y = fw1[(size_t)ncol * FD + k0 + kb + 1];
      acc = wmma4(a, b, acc);
    }
#pragma unroll
    for (int v = 0; v < 8; ++v) {
      int row = v + 8 * (lane >> 4);
      float f1v = f1ws[(size_t)(atom0 + row) * FD + ncol];
      gz1s[row * FD + ncol] = acc[v] * (1.0f - f1v * f1v);
    }
  }
  __syncthreads();

  // gDR = gz1 @ fw0^T     (16 x 1024)
  for (int nt = w * 8; nt < w * 8 + 8; ++nt) {
    const int pcol = nt * 16 + m;
    v8f acc = {0.f, 0.f, 0.f, 0.f, 0.f, 0.f, 0.f, 0.f};
    for (int k0 = 0; k0 < FD; k0 += 4) {
      v2f a, b;
      a.x = gz1s[m * FD + k0 + kb];
      a.y = gz1s[m * FD + k0 + kb + 1];
      b.x = fw0[(size_t)pcol * FD + k0 + kb];      // fw0^T
      b.y = fw0[(size_t)pcol * FD + k0 + kb + 1];
      acc = wmma4(a, b, acc);
    }
#pragma unroll
    for (int v = 0; v < 8; ++v) {
      int row = v + 8 * (lane >> 4);
      gDRws[(size_t)(atom0 + row) * DRD + pcol] = acc[v];
    }
  }
}

// ---------------------------------------------------------------------------
// Kernel 4: descriptor backward.  One wave per atom, recompute intermediates.
// ---------------------------------------------------------------------------
__global__ __launch_bounds__(32) void embed_bwd(
    const float* __restrict__ imdR,
    const float* __restrict__ ew0, const float* __restrict__ eb0,
    const float* __restrict__ ew1, const float* __restrict__ eb1,
    const float* __restrict__ gDRws,
    float* __restrict__ dEws) {
  __shared__ float Ri[KK * 4];
  __shared__ float Gs[KK * E2D];
  __shared__ float tA[16 * 4];
  __shared__ float tB[4 * E2D];
  __shared__ float gd[16 * E2D];     // gDR tile
  __shared__ float gA[16 * 4];       // gtmpA
  __shared__ float Ms[4 * E2D];      // gtmpB + embedded gtmpA
  __shared__ float ew0s[E1D], eb0s[E1D], eb1s[E2D];
  __shared__ float ew1s[E1D * E2D];

  const int atom = blockIdx.x;
  const int lane = threadIdx.x;
  const float* base = imdR + (size_t)atom * KK * 4;

  for (int e = lane; e < E1D; e += 32) { ew0s[e] = ew0[e]; eb0s[e] = eb0[e]; }
  for (int e = lane; e < E2D; e += 32) eb1s[e] = eb1[e];
  for (int e = lane; e < E1D * E2D; e += 32) ew1s[e] = ew1[e];
  for (int e = lane; e < DRD; e += 32) gd[e] = gDRws[((size_t)atom << 10) + e];
  __syncthreads();

  // --- recompute forward intermediates ---
  for (int k = lane; k < KK; k += 32) {
    float x = base[k * 4 + 0], y = base[k * 4 + 1], z = base[k * 4 + 2];
    float nb = base[k * 4 + 3];
    bool mask = nb > 0.0f;
    float dR2 = x * x + y * y + z * z;
    float safe = (dR2 > 0.0f) ? dR2 : 1.0f;
    float S = 0.0f;
    if (mask && dR2 < 10.0f) S = 1.0f / safe;
    if (dR2 >= 10.0f && dR2 < 25.0f)
      S += 0.5f * cosf(PI_F * (dR2 - 10.0f) / 15.0f) + 0.5f;
    float coef = mask ? (S / dR2) : 0.0f;
    Ri[k * 4 + 0] = S;
    Ri[k * 4 + 1] = coef * x;
    Ri[k * 4 + 2] = coef * y;
    Ri[k * 4 + 3] = coef * z;

    float h[E1D];
#pragma unroll
    for (int j = 0; j < E1D; ++j) h[j] = tanhf(S * ew0s[j] + eb0s[j]);
    for (int i = 0; i < E2D; ++i) {
      float acc = eb1s[i];
#pragma unroll
      for (int j = 0; j < E1D; ++j) acc += h[j] * ew1s[j * E2D + i];
      Gs[k * E2D + i] = tanhf(acc);
    }
  }
  __syncthreads();

  for (int e = lane; e < 64; e += 32) {
    int m = e >> 2, c = e & 3;
    float acc = 0.0f;
    for (int k = 0; k < KK; ++k) acc += Gs[k * E2D + m] * Ri[k * 4 + c];
    tA[e] = acc;
  }
  for (int e = lane; e < 4 * E2D; e += 32) {
    int c = e >> 6, i = e & 63;
    float acc = 0.0f;
    for (int k = 0; k < KK; ++k) acc += Ri[k * 4 + c] * Gs[k * E2D + i];
    tB[e] = acc;
  }
  __syncthreads();

  // gtmpA[m][c] = sum_p gDR[m][p] * tmpB[c][p]
  for (int e = lane; e < 64; e += 32) {
    int m = e >> 2, c = e & 3;
    float acc = 0.0f;
    for (int p = 0; p < E2D; ++p) acc += gd[m * E2D + p] * tB[c * E2D + p];
    gA[e] = acc;
  }
  __syncthreads();
  // M[c][i] = gtmpB[c][i] + (i<16 ? gtmpA[i][c] : 0),
  // gtmpB[c][i] = sum_m tmpA[m][c] * gDR[m][i]
  for (int e = lane; e < 4 * E2D; e += 32) {
    int c = e >> 6, i = e & 63;
    float acc = (i < 16) ? gA[i * 4 + c] : 0.0f;
    for (int m = 0; m < 16; ++m) acc += tA[m * 4 + c] * gd[m * E2D + i];
    Ms[e] = acc;
  }
  __syncthreads();

  // --- per-neighbor chain rule ---
  for (int k = lane; k < KK; k += 32) {
    float x = base[k * 4 + 0], y = base[k * 4 + 1], z = base[k * 4 + 2];
    float nb = base[k * 4 + 3];
    bool mask = nb > 0.0f;
    float dR2 = x * x + y * y + z * z;
    float S = Ri[k * 4 + 0];
    float coef = mask ? (S / dR2) : 0.0f;

    float gh[E1D];
#pragma unroll
    for (int j = 0; j < E1D; ++j) gh[j] = 0.0f;
    float gR0 = 0.f, gR1 = 0.f, gR2 = 0.f, gR3 = 0.f;
    float r0 = Ri[k * 4 + 0], r1 = Ri[k * 4 + 1], r2 = Ri[k * 4 + 2], r3 = Ri[k * 4 + 3];

    for (int i = 0; i < E2D; ++i) {
      float Gv = Gs[k * E2D + i];
      float M0 = Ms[0 * E2D + i], M1 = Ms[1 * E2D + i];
      float M2 = Ms[2 * E2D + i], M3 = Ms[3 * E2D + i];
      float gG = r0 * M0 + r1 * M1 + r2 * M2 + r3 * M3;
      gR0 += Gv * M0; gR1 += Gv * M1; gR2 += Gv * M2; gR3 += Gv * M3;
      float gz = gG * (1.0f - Gv * Gv);
#pragma unroll
      for (int j = 0; j < E1D; ++j) gh[j] += gz * ew1s[j * E2D + i];
    }

    float gS = gR0;
#pragma unroll
    for (int j = 0; j < E1D; ++j) {
      float hv = tanhf(S * ew0s[j] + eb0s[j]);
      float gz0 = gh[j] * (1.0f - hv * hv);
      gS += gz0 * ew0s[j];
    }

    float gcoef = gR1 * x + gR2 * y + gR3 * z;
    float gx = coef * gR1, gy = coef * gR2, gzc = coef * gR3;
    float gdR2 = 0.0f;
    if (mask) {
      gS += gcoef / dR2;
      gdR2 -= gcoef * S / (dR2 * dR2);
    }
    float dSd = 0.0f;
    if (mask && dR2 < 10.0f) dSd -= 1.0f / (dR2 * dR2);
    if (dR2 >= 10.0f && dR2 < 25.0f)
      dSd -= (PI_F / 30.0f) * sinf(PI_F * (dR2 - 10.0f) / 15.0f);
    gdR2 += gS * dSd;
    gx += 2.0f * x * gdR2;
    gy += 2.0f * y * gdR2;
    gzc += 2.0f * z * gdR2;

    size_t o = ((size_t)atom * KK + k) * 3;
    dEws[o + 0] = gx;
    dEws[o + 1] = gy;
    dEws[o + 2] = gzc;
  }
}

// ---------------------------------------------------------------------------
// Kernel 5: force assembly (local -sum + neighbor gather).
// ---------------------------------------------------------------------------
__global__ __launch_bounds__(256) void force_k(
    const float* __restrict__ imdR, const float* __restrict__ dEws,
    float* __restrict__ out) {
  int a = blockIdx.x * 256 + threadIdx.x;
  if (a >= NATOM) return;
  int b = a >> 11;   // atom / NN
  const float* base = imdR + (size_t)a * KK * 4;
  float fx = 0.f, fy = 0.f, fz = 0.f;
  for (int k = 0; k < KK; ++k) {
    size_t o = ((size_t)a * KK + k) * 3;
    fx -= dEws[o + 0];
    fy -= dEws[o + 1];
    fz -= dEws[o + 2];
    float nb = base[k * 4 + 3];
    if (nb > 0.0f) {
      int idx = (int)nb - 1;
      idx = idx < 0 ? 0 : (idx > NN - 1 ? NN - 1 : idx);
      size_t og = (((size_t)(b * NN + idx)) * KK + k) * 3;
      fx += dEws[og + 0];
      fy += dEws[og + 1];
      fz += dEws[og + 2];
    }
  }
  out[2 + NATOM + a * 3 + 0] = fx;
  out[2 + NATOM + a * 3 + 1] = fy;
  out[2 + NATOM + a * 3 + 2] = fz;
}

// ---------------------------------------------------------------------------
extern "C" void kernel_launch(void* const* d_in, const int* in_sizes, int n_in,
                              void* d_out, int out_size, void* d_ws, size_t ws_size,
                              hipStream_t stream) {
  const float* imdR = (const float*)d_in[0];
  // d_in[1] = neighbor (redundant: same info as image_dR[...,3])
  const float* ew0 = (const float*)d_in[2];
  const float* eb0 = (const float*)d_in[3];
  const float* ew1 = (const float*)d_in[4];
  const float* eb1 = (const float*)d_in[5];
  const float* fw0 = (const float*)d_in[6];
  const float* fb0 = (const float*)d_in[7];
  const float* fw1 = (const float*)d_in[8];
  const float* fb1 = (const float*)d_in[9];
  const float* fw2 = (const float*)d_in[10];
  const float* fb2 = (const float*)d_in[11];
  float* out = (float*)d_out;

  float* ws = (float*)d_ws;
  float* DRws  = ws;                               // NATOM * 1024
  float* f1ws  = DRws  + (size_t)NATOM * DRD;      // NATOM * 256
  float* f2ws  = f1ws  + (size_t)NATOM * FD;       // NATOM * 256
  float* gDRws = f2ws  + (size_t)NATOM * FD;       // NATOM * 1024
  float* dEws  = gDRws + (size_t)NATOM * DRD;      // NATOM * 96 * 3

  hipMemsetAsync(out, 0, 2 * sizeof(float), stream);   // Etot accumulators

  embed_fwd<<<NATOM, 32, 0, stream>>>(imdR, ew0, eb0, ew1, eb1, DRws);
  fit_fwd<<<NATOM / 16, 256, 0, stream>>>(DRws, fw0, fb0, fw1, fb1, fw2, fb2,
                                          f1ws, f2ws, out);
  fit_bwd<<<NATOM / 16, 256, 0, stream>>>(f1ws, f2ws, fw0, fw1, fw2, gDRws);
  embed_bwd<<<NATOM, 32, 0, stream>>>(imdR, ew0, eb0, ew1, eb1, gDRws, dEws);
  force_k<<<NATOM / 256, 256, 0, stream>>>(imdR, dEws, out);
}